// Inverse_13572096655809
// MI455X (gfx1250) — compile-verified
//
#include <hip/hip_runtime.h>
#include <stdint.h>

// ---------------- CDNA5 async global<->LDS data movement ----------------
typedef int v4i __attribute__((ext_vector_type(4)));
typedef __attribute__((address_space(1))) v4i GV4;   // global int4
typedef __attribute__((address_space(3))) v4i LV4;   // LDS int4

#if __has_builtin(__builtin_amdgcn_global_load_async_to_lds_b128)
#define ASYNC_LD_B128(gp, lp, off)                                   \
    __builtin_amdgcn_global_load_async_to_lds_b128(                  \
        (GV4*)(uintptr_t)(gp), (LV4*)(uintptr_t)(lp), (off), 0)
#else
#define ASYNC_LD_B128(gp, lp, off)                                              \
    asm volatile("global_load_async_to_lds_b128 %0, %1, off offset:%2"          \
                 :: "v"((unsigned)(uintptr_t)(lp)),                             \
                    "v"((unsigned long long)(uintptr_t)(gp)),                   \
                    "i"(off) : "memory")
#endif

#if __has_builtin(__builtin_amdgcn_global_store_async_from_lds_b128)
#define ASYNC_ST_B128(gp, lp, off)                                   \
    __builtin_amdgcn_global_store_async_from_lds_b128(               \
        (GV4*)(uintptr_t)(gp), (LV4*)(uintptr_t)(lp), (off), 0)
#else
#define ASYNC_ST_B128(gp, lp, off)                                              \
    asm volatile("global_store_async_from_lds_b128 %0, %1, off offset:%2"       \
                 :: "v"((unsigned long long)(uintptr_t)(gp)),                   \
                    "v"((unsigned)(uintptr_t)(lp)),                             \
                    "i"(off) : "memory")
#endif

#if __has_builtin(__builtin_amdgcn_s_wait_asynccnt)
#define WAIT_ASYNC0() __builtin_amdgcn_s_wait_asynccnt(0)
#else
#define WAIT_ASYNC0() asm volatile("s_wait_asynccnt 0" ::: "memory")
#endif

// ---------------- tiling ----------------
#define THREADS        256
#define TILE_ROWS      1024                       // rows per block
#define ROWS_PER_THR   (TILE_ROWS / THREADS)      // 4
#define TILE_FLOATS    (TILE_ROWS * 9)            // 9216
#define TILE_BYTES     (TILE_FLOATS * 4)          // 36864 (16B-aligned per tile)
#define CHUNK_BYTES    (THREADS * 16)             // 4096 per async chunk
#define NCHUNKS        (TILE_BYTES / CHUNK_BYTES) // 9 exactly

#define DO9(M) M(0) M(1) M(2) M(3) M(4) M(5) M(6) M(7) M(8)

// rcp + 1 Newton step (~0.5 ulp); inputs well away from denormal range
__device__ __forceinline__ float rcp_nr(float x) {
    float r = __builtin_amdgcn_rcpf(x);
    return r * __builtin_fmaf(-x, r, 2.0f);
}

// inverse of upper-triangular [a b c; 0 d e; 0 0 f], row-major 9-vector out
__device__ __forceinline__ void inv_upper3(float a, float b, float c,
                                           float d, float e, float f,
                                           float o[9]) {
    const float ia = rcp_nr(a);
    const float id = rcp_nr(d);
    const float iff = rcp_nr(f);
    const float iad = ia * id;
    o[0] = ia;
    o[1] = -b * iad;
    o[2] = __builtin_fmaf(b, e, -(c * d)) * (iad * iff);
    o[3] = 0.0f;
    o[4] = id;
    o[5] = -e * (id * iff);
    o[6] = 0.0f;
    o[7] = 0.0f;
    o[8] = iff;
}

__global__ __launch_bounds__(THREADS)
void inv3x3_async_kernel(const float* __restrict__ x, float* __restrict__ out, int nrows) {
    __shared__ __align__(16) float tile[TILE_FLOATS];   // staged in place: in -> out

    const int tid = threadIdx.x;
    const long long tileRow  = (long long)blockIdx.x * TILE_ROWS;
    const long long tileByte = tileRow * 36;

    if (tileRow + TILE_ROWS <= (long long)nrows) {
        // ---- async stage-in: 9 x 4KB chunks, 16B per lane, fully coalesced ----
        const char* gin = (const char*)x + tileByte + (long long)tid * 16;
        char*       lin = (char*)tile + tid * 16;
#define LD_CHUNK(K) ASYNC_LD_B128(gin, lin, (K) * CHUNK_BYTES);
        DO9(LD_CHUNK)
#undef LD_CHUNK
        WAIT_ASYNC0();
        __syncthreads();

        // ---- compute 4 rows per thread, in place in LDS (bank-conflict-free) ----
#pragma unroll
        for (int r = 0; r < ROWS_PER_THR; ++r) {
            const int row = tid + r * THREADS;
            float* p = &tile[row * 9];
            const float a = p[0], b = p[1], c = p[2];
            const float d = p[4], e = p[5];
            const float f = p[8];
            float o[9];
            inv_upper3(a, b, c, d, e, f, o);
#pragma unroll
            for (int j = 0; j < 9; ++j) p[j] = o[j];
        }
        __syncthreads();

        // ---- async stage-out: LDS -> global, same chunking ----
        char* gout = (char*)out + tileByte + (long long)tid * 16;
        char* lout = (char*)tile + tid * 16;
#define ST_CHUNK(K) ASYNC_ST_B128(gout, lout, (K) * CHUNK_BYTES);
        DO9(ST_CHUNK)
#undef ST_CHUNK
        // S_ENDPGM performs an implicit wait-idle, draining ASYNCcnt.
    } else {
        // ---- tail tile (not hit for N = 8388608, kept for generality) ----
#pragma unroll
        for (int r = 0; r < ROWS_PER_THR; ++r) {
            const long long row = tileRow + tid + r * THREADS;
            if (row < (long long)nrows) {
                const float* p = x + row * 9;
                float o[9];
                inv_upper3(p[0], p[1], p[2], p[4], p[5], p[8], o);
                float* q = out + row * 9;
#pragma unroll
                for (int j = 0; j < 9; ++j) q[j] = o[j];
            }
        }
    }
}

extern "C" void kernel_launch(void* const* d_in, const int* in_sizes, int n_in,
                              void* d_out, int out_size, void* d_ws, size_t ws_size,
                              hipStream_t stream) {
    const float* x = (const float*)d_in[0];
    float* out = (float*)d_out;
    const int nrows = in_sizes[0] / 9;
    const int blocks = (nrows + TILE_ROWS - 1) / TILE_ROWS;
    hipLaunchKernelGGL(inv3x3_async_kernel, dim3(blocks), dim3(THREADS), 0, stream,
                       x, out, nrows);
}